// ImplicitBoomerang_61125974557357
// MI455X (gfx1250) — compile-verified
//
#include <hip/hip_runtime.h>
#include <math.h>

// ---------------------------------------------------------------------------
// ImplicitBoomerang on MI455X (gfx1250, wave32)
//
// Strategy:
//  * Precompute C = x @ [Wl_x;Wu_x;Wf_x]^T + biases once (reads 32MB x once).
//    WMMA f32 16x16x4 GEMM, M=8192 K=1024 N=64.
//  * 20 gated iterations, each L2-resident (~4MB working set):
//      conv-check (latches done flag) -> lam/u (scalar dot32, LDS weights)
//      -> channel-major work-efficient scan -> f_theta elementwise update.
//  * Final lam_u + scan (forced), then output GEMM (WMMA f32 16x16x4,
//    M=8192 N=1024 K=48) writing the 32MB result.
// HBM floor ~64MB => ~3us at 23.3 TB/s; everything else lives in L2.
// ---------------------------------------------------------------------------

#define SEQ 8192
#define DM 1024
#define DS 16
#define DI 32
#define MAX_ITERS 20
#define DT_C 0.001f
#define RHO_C 30.0f
#define BETA_C 15.6f
#define GAMMA_C 0.26f

typedef __attribute__((ext_vector_type(2))) float v2f;
typedef __attribute__((ext_vector_type(8))) float v8f;

__device__ __forceinline__ float sigmoidf_(float x) { return 1.0f / (1.0f + expf(-x)); }

// --------------------------------------------------------------------------
// init: z = 0, z_prev = +inf, done = 0  (must run every call; ws not re-poisoned)
// --------------------------------------------------------------------------
__global__ void boom_init_kernel(float* __restrict__ z, float* __restrict__ zp,
                                 int* __restrict__ done) {
  int i = blockIdx.x * blockDim.x + threadIdx.x;
  if (i < SEQ * DI) {
    z[i]  = 0.0f;
    zp[i] = __builtin_huge_valf();
  }
  if (i == 0) *done = 0;
}

// --------------------------------------------------------------------------
// Precompute C[s][0:16)=x@Wl_x^T+bl, [16:32)=x@Wu_x^T+bu, [32:64)=x@Wf_x^T+bf
// WMMA f32 16x16x4. One block per 16-row M tile; 4 waves cover the 64 N cols.
// A layout (ISA 7.12.2): lanes 0-15 M=lane K={0,1}; lanes 16-31 M=lane-16 K={2,3}.
// B layout assumed mirror: lanes 0-15 N=lane K={0,1}; lanes 16-31 K={2,3}.
// D: VGPR r -> (M = half*8 + r, N = lane&15).
// --------------------------------------------------------------------------
__global__ void boom_precomp_kernel(const float* __restrict__ x,
                                    const float* __restrict__ Wl, const float* __restrict__ bl,
                                    const float* __restrict__ Wu, const float* __restrict__ bu,
                                    const float* __restrict__ Wf, const float* __restrict__ bf,
                                    float* __restrict__ C) {
  const int lane = threadIdx.x & 31;
  const int wave = threadIdx.x >> 5;
  const int m0   = blockIdx.x * 16;

  const float* W; const float* bias; int ld, koff, colbase, nbase;
  if (wave == 0)      { W = Wl; bias = bl; ld = DI + DM; koff = DI; colbase = 0;  nbase = 0;  }
  else if (wave == 1) { W = Wu; bias = bu; ld = DI + DM; koff = DI; colbase = 16; nbase = 0;  }
  else if (wave == 2) { W = Wf; bias = bf; ld = DS + DM; koff = DS; colbase = 32; nbase = 0;  }
  else                { W = Wf; bias = bf; ld = DS + DM; koff = DS; colbase = 48; nbase = 16; }

  const int half = lane >> 4;       // 0 or 1
  const int l16  = lane & 15;
  const int kk   = half * 2;        // K sub-offset for this lane's A/B fragment
  const float* xr = x + (size_t)(m0 + l16) * DM;
  const float* wr = W + (size_t)(nbase + l16) * ld + koff;

  v8f acc = {};
  #pragma unroll 8
  for (int k = 0; k < DM; k += 4) {
    v2f a, b;
    a.x = xr[k + kk];     a.y = xr[k + kk + 1];
    b.x = wr[k + kk];     b.y = wr[k + kk + 1];
    acc = __builtin_amdgcn_wmma_f32_16x16x4_f32(false, a, false, b, (short)0, acc,
                                                false, false);
  }
  const float bv = bias[nbase + l16];
  #pragma unroll
  for (int r = 0; r < 8; ++r) {
    int row = m0 + half * 8 + r;
    C[(size_t)row * 64 + colbase + l16] = acc[r] + bv;
  }
}

// --------------------------------------------------------------------------
// Convergence check: done |= (||z - zp|| / ||zp|| < 1e-3).  Single block.
// First iteration: zp = inf -> ratio = nan -> stays not-done (matches JAX).
// --------------------------------------------------------------------------
__global__ void boom_conv_kernel(const float* __restrict__ z, const float* __restrict__ zp,
                                 int* __restrict__ done) {
  __shared__ float sn[1024];
  __shared__ float sd[1024];
  const int tid = threadIdx.x;
  float num = 0.0f, den = 0.0f;
  for (int i = tid; i < SEQ * DI; i += 1024) {
    float d = z[i] - zp[i];
    num += d * d;
    den += zp[i] * zp[i];
  }
  sn[tid] = num; sd[tid] = den;
  __syncthreads();
  for (int st = 512; st > 0; st >>= 1) {
    if (tid < st) { sn[tid] += sn[tid + st]; sd[tid] += sd[tid + st]; }
    __syncthreads();
  }
  if (tid == 0) {
    float r = sqrtf(sn[0]) / sqrtf(sd[0]);
    if (r < 0.001f) *done = 1;
  }
}

// --------------------------------------------------------------------------
// lam = sigmoid(z @ Wl_z^T + Cl), u = z @ Wu_z^T + Cu.  K=32 dot per output,
// weights staged in LDS. Outputs channel-major [ch][s] for the scan kernel.
// --------------------------------------------------------------------------
__global__ void boom_lamu_kernel(const float* __restrict__ z,
                                 const float* __restrict__ Wl, const float* __restrict__ Wu,
                                 const float* __restrict__ C,
                                 float* __restrict__ lam, float* __restrict__ u,
                                 const int* __restrict__ done, int force) {
  if (!force && *done) return;
  __shared__ float swl[DS][DI];
  __shared__ float swu[DS][DI];
  const int tid = threadIdx.x;
  for (int idx = tid; idx < DS * DI; idx += 256) {
    int o = idx >> 5, c = idx & 31;
    swl[o][c] = Wl[o * (DI + DM) + c];
    swu[o][c] = Wu[o * (DI + DM) + c];
  }
  __syncthreads();
  const int row = blockIdx.x * 256 + tid;
  float zr[DI];
  #pragma unroll
  for (int c = 0; c < DI; ++c) zr[c] = z[(size_t)row * DI + c];
  const float* crow = C + (size_t)row * 64;
  for (int o = 0; o < DS; ++o) {
    float sl = crow[o], su = crow[16 + o];
    #pragma unroll
    for (int c = 0; c < DI; ++c) { sl += zr[c] * swl[o][c]; su += zr[c] * swu[o][c]; }
    lam[(size_t)o * SEQ + row] = sigmoidf_(sl);
    u  [(size_t)o * SEQ + row] = su;
  }
}

// --------------------------------------------------------------------------
// Linear recurrence h_t = lam_t*h_{t-1} + u_t, h_{-1}=0, 16 channels.
// One block, 16 waves (one per channel).  Work-efficient: each lane serially
// reduces a contiguous 256-elem chunk (float4 loads), wave-scans the 32
// aggregates with the associative combine, then re-applies.  Critical path
// ~O(512) FMAs, all lanes busy.
// --------------------------------------------------------------------------
__global__ void boom_scan_kernel(const float* __restrict__ lam, const float* __restrict__ u,
                                 float* __restrict__ h,
                                 const int* __restrict__ done, int force) {
  if (!force && *done) return;
  const int lane = threadIdx.x & 31;
  const int ch   = threadIdx.x >> 5;     // 0..15
  const float* lch = lam + (size_t)ch * SEQ;
  const float* uch = u   + (size_t)ch * SEQ;
  float*       hch = h   + (size_t)ch * SEQ;
  const int s0 = lane * 256;

  // Pass 1: chunk aggregate (Lc, Uc) = fold of combine over 256 elements.
  float Lc = 1.0f, Uc = 0.0f;
  for (int i = 0; i < 256; i += 4) {
    const float4 l4 = *(const float4*)(lch + s0 + i);
    const float4 u4 = *(const float4*)(uch + s0 + i);
    Uc = l4.x * Uc + u4.x;  Lc *= l4.x;
    Uc = l4.y * Uc + u4.y;  Lc *= l4.y;
    Uc = l4.z * Uc + u4.z;  Lc *= l4.z;
    Uc = l4.w * Uc + u4.w;  Lc *= l4.w;
  }
  // Inclusive wave scan of aggregates with combine(a,b) = (la*lb, lb*ua+ub).
  float Li = Lc, Ui = Uc;
  #pragma unroll
  for (int off = 1; off < 32; off <<= 1) {
    float Lp = __shfl_up(Li, off, 32);
    float Up = __shfl_up(Ui, off, 32);
    if (lane >= off) { Ui = Li * Up + Ui; Li = Li * Lp; }
  }
  // Exclusive prefix: h-value entering this lane's chunk.
  float Uex = __shfl_up(Ui, 1, 32);
  if (lane == 0) Uex = 0.0f;

  // Pass 2: replay chunk with carry, emit h.
  float carry = Uex;
  for (int i = 0; i < 256; i += 4) {
    const float4 l4 = *(const float4*)(lch + s0 + i);
    const float4 u4 = *(const float4*)(uch + s0 + i);
    float4 hv;
    carry = l4.x * carry + u4.x;  hv.x = carry;
    carry = l4.y * carry + u4.y;  hv.y = carry;
    carry = l4.z * carry + u4.z;  hv.z = carry;
    carry = l4.w * carry + u4.w;  hv.w = carry;
    *(float4*)(hch + s0 + i) = hv;
  }
}

// --------------------------------------------------------------------------
// f_theta: a = h_prev @ Wf_h^T + Cf; alpha/sigma sigmoids; ODE step.
// Writes zp = old z, z = updated z.  Gated on done (matches jnp.where freeze).
// --------------------------------------------------------------------------
__global__ void boom_f_kernel(const float* __restrict__ h, const float* __restrict__ Wf,
                              const float* __restrict__ C,
                              float* __restrict__ z, float* __restrict__ zp,
                              const int* __restrict__ done) {
  if (*done) return;
  __shared__ float swf[DI][DS];
  const int tid = threadIdx.x;
  for (int idx = tid; idx < DI * DS; idx += 256) {
    int o = idx >> 4, c = idx & 15;
    swf[o][c] = Wf[o * (DS + DM) + c];
  }
  __syncthreads();
  const int row = blockIdx.x * 256 + tid;

  float hp[DS];
  if (row == 0) {
    #pragma unroll
    for (int c = 0; c < DS; ++c) hp[c] = 0.0f;
  } else {
    #pragma unroll
    for (int c = 0; c < DS; ++c) hp[c] = h[(size_t)c * SEQ + (row - 1)];
  }
  const float* crow = C + (size_t)row * 64 + 32;
  float a[DI];
  for (int o = 0; o < DI; ++o) {
    float s = crow[o];
    #pragma unroll
    for (int c = 0; c < DS; ++c) s += hp[c] * swf[o][c];
    a[o] = s;
  }
  float* zr  = z  + (size_t)row * DI;
  float* zpr = zp + (size_t)row * DI;
  #pragma unroll
  for (int c = 0; c < DS; ++c) {
    float uo = zr[c], vo = zr[16 + c];
    float alpha = sigmoidf_(a[c]);
    float sigma = sigmoidf_(a[16 + c]);
    float th = tanhf(RHO_C * (vo - uo));
    float du =  1.0f - alpha * expf(BETA_C * vo) * (1.0f - GAMMA_C * (0.3f - uo)) + sigma * th;
    float dv = -1.0f + alpha * expf(BETA_C * uo) * (1.0f + GAMMA_C * (0.3f - vo)) + sigma * th;
    zpr[c]      = uo;
    zpr[16 + c] = vo;
    zr[c]       = uo + DT_C * du;
    zr[16 + c]  = vo + DT_C * dv;
  }
}

// --------------------------------------------------------------------------
// out = [z | h] @ Wo^T + bo.  WMMA f32 16x16x4, M=8192 N=1024 K=48.
// grid (512 M-tiles, 8), 8 waves/block -> 64 N-tiles of 16.
// --------------------------------------------------------------------------
__global__ void boom_out_kernel(const float* __restrict__ z, const float* __restrict__ h,
                                const float* __restrict__ Wo, const float* __restrict__ bo,
                                float* __restrict__ out) {
  const int lane  = threadIdx.x & 31;
  const int wave  = threadIdx.x >> 5;
  const int m0    = blockIdx.x * 16;
  const int n0    = (blockIdx.y * 8 + wave) * 16;
  const int half  = lane >> 4;
  const int l16   = lane & 15;
  const int kk    = half * 2;
  const int arow  = m0 + l16;
  const int ncol  = n0 + l16;
  const float* zrow = z  + (size_t)arow * DI;
  const float* wrow = Wo + (size_t)ncol * (DI + DS);

  v8f acc = {};
  #pragma unroll
  for (int k = 0; k < DI + DS; k += 4) {
    const int ka = k + kk;                      // kk in {0,2}; DI%4==0 => no mixed step
    v2f a, b;
    if (ka < DI) { a.x = zrow[ka];  a.y = zrow[ka + 1]; }
    else         { a.x = h[(size_t)(ka - DI) * SEQ + arow];
                   a.y = h[(size_t)(ka + 1 - DI) * SEQ + arow]; }
    b.x = wrow[ka]; b.y = wrow[ka + 1];
    acc = __builtin_amdgcn_wmma_f32_16x16x4_f32(false, a, false, b, (short)0, acc,
                                                false, false);
  }
  const float bv = bo[ncol];
  #pragma unroll
  for (int r = 0; r < 8; ++r) {
    int row = m0 + half * 8 + r;
    out[(size_t)row * DM + ncol] = acc[r] + bv;
  }
}

// --------------------------------------------------------------------------
extern "C" void kernel_launch(void* const* d_in, const int* in_sizes, int n_in,
                              void* d_out, int out_size, void* d_ws, size_t ws_size,
                              hipStream_t stream) {
  const float* x  = (const float*)d_in[0];
  const float* Wf = (const float*)d_in[1];
  const float* bf = (const float*)d_in[2];
  const float* Wl = (const float*)d_in[3];
  const float* bl = (const float*)d_in[4];
  const float* Wu = (const float*)d_in[5];
  const float* bu = (const float*)d_in[6];
  const float* Wo = (const float*)d_in[7];
  const float* bo = (const float*)d_in[8];
  float* out = (float*)d_out;

  float* ws  = (float*)d_ws;
  float* C   = ws;  ws += (size_t)SEQ * 64;   // 2.0 MB  [s][64]: Cl|Cu|Cf
  float* z   = ws;  ws += (size_t)SEQ * DI;   // 1.0 MB
  float* zp  = ws;  ws += (size_t)SEQ * DI;   // 1.0 MB
  float* lam = ws;  ws += (size_t)SEQ * DS;   // 0.5 MB  channel-major [ch][s]
  float* u   = ws;  ws += (size_t)SEQ * DS;   // 0.5 MB  channel-major
  float* h   = ws;  ws += (size_t)SEQ * DS;   // 0.5 MB  channel-major
  int*   done = (int*)ws;                     // 4 B

  boom_init_kernel<<<(SEQ * DI + 255) / 256, 256, 0, stream>>>(z, zp, done);
  boom_precomp_kernel<<<SEQ / 16, 128, 0, stream>>>(x, Wl, bl, Wu, bu, Wf, bf, C);

  for (int it = 0; it < MAX_ITERS; ++it) {
    boom_conv_kernel<<<1, 1024, 0, stream>>>(z, zp, done);
    boom_lamu_kernel<<<SEQ / 256, 256, 0, stream>>>(z, Wl, Wu, C, lam, u, done, 0);
    boom_scan_kernel<<<1, 512, 0, stream>>>(lam, u, h, done, 0);
    boom_f_kernel<<<SEQ / 256, 256, 0, stream>>>(h, Wf, C, z, zp, done);
  }

  // Final h* from converged z* (forced: done flag may be set).
  boom_lamu_kernel<<<SEQ / 256, 256, 0, stream>>>(z, Wl, Wu, C, lam, u, done, 1);
  boom_scan_kernel<<<1, 512, 0, stream>>>(lam, u, h, done, 1);
  boom_out_kernel<<<dim3(SEQ / 16, 8), 256, 0, stream>>>(z, h, Wo, bo, out);
}